// DualAttributionModel_17506286698701
// MI455X (gfx1250) — compile-verified
//
#include <hip/hip_runtime.h>
#include <hip/hip_bf16.h>

#define EPSV 1e-5f

typedef float v2f __attribute__((ext_vector_type(2)));
typedef float v8f __attribute__((ext_vector_type(8)));

// ---------------------------------------------------------------- utilities
__global__ void k_zero(float* __restrict__ p, int n) {
  int i = blockIdx.x * blockDim.x + threadIdx.x;
  int stride = gridDim.x * blockDim.x;
  for (; i < n; i += stride) p[i] = 0.f;
}

__global__ void k_degree(const int* __restrict__ dst, float* __restrict__ deg, int E) {
  int i = blockIdx.x * blockDim.x + threadIdx.x;
  if (i < E) unsafeAtomicAdd(&deg[dst[i]], 1.0f);
}

__global__ void k_dinv(float* __restrict__ d, int n) {
  int i = blockIdx.x * blockDim.x + threadIdx.x;
  if (i < n) d[i] = rsqrtf(d[i] + 1.0f);  // +1 self-loop => deg >= 1 always
}

// ------------------------------------------------ fp32 WMMA GEMM (16x16x4)
// C = A(MxK) @ B(KxNcols) [+ bias], row-major. M must be a multiple of 16.
// One wave32 per 16x16 output tile. Operand layout per CDNA5 ISA 7.12.2:
//   A 16x4 : lanes 0-15 hold {K=0,K=1}, lanes 16-31 hold {K=2,K=3}, M = lane&15
//   B 4x16 : same K split, N = lane&15
//   C/D    : VGPR v -> row v + 8*(lane>=16), col = lane&15
// Hot loop is branch-free: ragged columns handled by clamped address + 0/1
// mask multiply; ragged K handled in a single tail step outside the loop.
__global__ void k_gemm_wmma(const float* __restrict__ A, const float* __restrict__ B,
                            const float* __restrict__ bias, float* __restrict__ C,
                            int M, int K, int Ncols, int tilesN) {
  int wave = blockIdx.x * (blockDim.x >> 5) + (threadIdx.x >> 5);
  int totalTiles = (M >> 4) * tilesN;
  if (wave >= totalTiles) return;                 // wave-uniform exit
  int tm = wave / tilesN, tn = wave - tm * tilesN;
  int lane = threadIdx.x & 31;
  int half = lane >> 4, l16 = lane & 15;
  int arow = (tm << 4) + l16;
  int col  = (tn << 4) + l16;
  bool colv = col < Ncols;
  int colc = colv ? col : (Ncols - 1);            // safe clamped address
  float bmask = colv ? 1.f : 0.f;                 // zero out-of-range B lanes
  const float* Ap = A + (size_t)arow * K + (half << 1);
  const float* Bp = B + colc + (size_t)(half << 1) * Ncols;
  const size_t ldb = (size_t)Ncols;
  int K4 = K & ~3;
  v8f acc = {};
#pragma unroll 4
  for (int k0 = 0; k0 < K4; k0 += 4) {
    v2f av, bv;
    av.x = Ap[k0];
    av.y = Ap[k0 + 1];
    bv.x = Bp[(size_t)k0 * ldb] * bmask;
    bv.y = Bp[(size_t)(k0 + 1) * ldb] * bmask;
    acc = __builtin_amdgcn_wmma_f32_16x16x4_f32(false, av, false, bv,
                                                (short)0, acc, false, false);
  }
  if (K4 < K) {                                   // single ragged-K step
    int ka = K4 + (half << 1);
    float ax = 0.f, ay = 0.f, bx = 0.f, by = 0.f;
    if (ka < K) {
      ax = Ap[K4];
      bx = Bp[(size_t)K4 * ldb] * bmask;
    }
    if (ka + 1 < K) {
      ay = Ap[K4 + 1];
      by = Bp[(size_t)(K4 + 1) * ldb] * bmask;
    }
    v2f av = {ax, ay};
    v2f bv = {bx, by};
    acc = __builtin_amdgcn_wmma_f32_16x16x4_f32(false, av, false, bv,
                                                (short)0, acc, false, false);
  }
  if (colv) {
    float bb = bias ? bias[col] : 0.f;
    int r0 = (tm << 4) + (half << 3);
#pragma unroll
    for (int v = 0; v < 8; ++v)
      C[(size_t)(r0 + v) * Ncols + col] = acc[v] + bb;
  }
}

// ------------------------------------------------ edge scatter (C == 128)
// out[dst] += h[src] * dinv[src]*dinv[dst]; one wave per edge, float4 / lane.
__global__ void k_scatter128(const float* __restrict__ h, const int* __restrict__ src,
                             const int* __restrict__ dst, const float* __restrict__ dinv,
                             float* __restrict__ out, int E) {
  int nw   = (gridDim.x * blockDim.x) >> 5;
  int w    = (blockIdx.x * blockDim.x + threadIdx.x) >> 5;
  int lane = threadIdx.x & 31;
  for (int e = w; e < E; e += nw) {
    int s = src[e], d = dst[e];
    float coef = dinv[s] * dinv[d];
    const float4* hp = (const float4*)(h + ((size_t)s << 7));
    float4 m = hp[lane];
    float* op = out + ((size_t)d << 7) + (lane << 2);
    unsafeAtomicAdd(op + 0, m.x * coef);
    unsafeAtomicAdd(op + 1, m.y * coef);
    unsafeAtomicAdd(op + 2, m.z * coef);
    unsafeAtomicAdd(op + 3, m.w * coef);
    int e2 = e + nw;                         // prefetch next edge's source row
    if (e2 < E) {
      int s2 = src[e2];
      __builtin_prefetch(h + ((size_t)s2 << 7) + (lane << 2), 0, 3);
    }
  }
}

// t1 += t0 * dinv[i]^2 + b[c]   (self-loop contribution + conv bias)
// C == 1<<shC
__global__ void k_finalize(float* __restrict__ t1, const float* __restrict__ t0,
                           const float* __restrict__ dinv, const float* __restrict__ b,
                           int n, int shC) {
  int i = blockIdx.x * blockDim.x + threadIdx.x;
  int total = n << shC;
  int stride = gridDim.x * blockDim.x;
  int maskC = (1 << shC) - 1;
  for (; i < total; i += stride) {
    int r = i >> shC, c = i & maskC;
    float dv = dinv[r];
    t1[i] = t1[i] + t0[i] * dv * dv + b[c];
  }
}

// ------------------------------------------------ BatchNorm (train stats)
__global__ void k_bn_stats(const float* __restrict__ X, float* __restrict__ stats,
                           int n, int C, int rpb) {
  int c = threadIdx.x;                       // blockDim.x == C
  int r0 = blockIdx.x * rpb;
  int r1 = r0 + rpb; if (r1 > n) r1 = n;
  float s = 0.f, s2 = 0.f;
  for (int r = r0; r < r1; ++r) {
    float v = X[(size_t)r * C + c];
    s += v; s2 += v * v;
  }
  unsafeAtomicAdd(&stats[c], s);
  unsafeAtomicAdd(&stats[C + c], s2);
}

// C == 1<<shC
__global__ void k_bn_apply(const float* __restrict__ X, const float* __restrict__ stats,
                           const float* __restrict__ g, const float* __restrict__ be,
                           float* __restrict__ Y, int n, int shC, int relu) {
  int i = blockIdx.x * blockDim.x + threadIdx.x;
  int total = n << shC;
  int stride = gridDim.x * blockDim.x;
  int C = 1 << shC, maskC = C - 1;
  float inv = 1.f / (float)n;
  for (; i < total; i += stride) {
    int c = i & maskC;
    float m   = stats[c] * inv;
    float var = stats[C + c] * inv - m * m;   // biased variance
    float y = (X[i] - m) * rsqrtf(var + EPSV) * g[c] + be[c];
    if (relu) y = fmaxf(y, 0.f);
    Y[i] = y;
  }
}

// ---------------------------------------------------------------- launcher
extern "C" void kernel_launch(void* const* d_in, const int* in_sizes, int n_in,
                              void* d_out, int out_size, void* d_ws, size_t ws_size,
                              hipStream_t stream) {
  (void)n_in; (void)out_size; (void)ws_size;
  const int IN = 165, H = 128, H2 = 64, NEX = 50, NCAT = 3;
  const int SH = 7, SH2 = 6;               // log2(H), log2(H2)

  const float* x   = (const float*)d_in[0];
  const int*   ei  = (const int*)  d_in[1];
  const float* W1  = (const float*)d_in[2];
  const float* b1  = (const float*)d_in[3];
  const float* W2  = (const float*)d_in[4];
  const float* b2  = (const float*)d_in[5];
  const float* g1  = (const float*)d_in[6];
  const float* be1 = (const float*)d_in[7];
  const float* g2  = (const float*)d_in[8];
  const float* be2 = (const float*)d_in[9];
  const float* fW1 = (const float*)d_in[10];
  const float* fb1 = (const float*)d_in[11];
  const float* fg1 = (const float*)d_in[12];
  const float* fbe1= (const float*)d_in[13];
  const float* fW2 = (const float*)d_in[14];
  const float* fb2 = (const float*)d_in[15];
  const float* fg2 = (const float*)d_in[16];
  const float* fbe2= (const float*)d_in[17];
  const float* fW3 = (const float*)d_in[18];
  const float* fb3 = (const float*)d_in[19];
  const float* bW1 = (const float*)d_in[20];
  const float* bb1 = (const float*)d_in[21];
  const float* bg1 = (const float*)d_in[22];
  const float* bbe1= (const float*)d_in[23];
  const float* bW2 = (const float*)d_in[24];
  const float* bb2 = (const float*)d_in[25];

  const int N = in_sizes[0] / IN;
  const int E = in_sizes[1] / 2;
  const int* src = ei;
  const int* dst = ei + E;

  float* ws    = (float*)d_ws;
  float* dinv  = ws;                       // N
  float* stats = ws + N;                   // 256
  float* t0 = stats + 256;                 // N*H
  float* t1 = t0 + (size_t)N * H;          // N*H
  float* t2 = t1 + (size_t)N * H;          // N*H2

  float* outp = (float*)d_out;
  float* fwd = outp;                       // N*NEX
  float* bwd = outp + (size_t)N * NEX;     // N*NCAT
  float* emb = bwd  + (size_t)N * NCAT;    // N*H

  auto cdiv = [](int a, int b) { return (a + b - 1) / b; };
  const int TPB = 256;                     // 8 wave32 per block

  // ---- degrees -> dinv
  k_zero  <<<cdiv(N, TPB), TPB, 0, stream>>>(dinv, N);
  k_degree<<<cdiv(E, TPB), TPB, 0, stream>>>(dst, dinv, E);
  k_dinv  <<<cdiv(N, TPB), TPB, 0, stream>>>(dinv, N);

  // ---- GCN layer 1: t0 = x @ W1 ; aggregate -> t1 ; BN+ReLU in place
  { int tn = cdiv(H, 16), tiles = (N / 16) * tn;
    k_gemm_wmma<<<cdiv(tiles, 8), TPB, 0, stream>>>(x, W1, nullptr, t0, N, IN, H, tn); }
  k_zero      <<<cdiv(N * H, TPB), TPB, 0, stream>>>(t1, N * H);
  k_scatter128<<<4096, TPB, 0, stream>>>(t0, src, dst, dinv, t1, E);
  k_finalize  <<<cdiv(N * H, TPB), TPB, 0, stream>>>(t1, t0, dinv, b1, N, SH);
  k_zero      <<<1, 256, 0, stream>>>(stats, 2 * H);
  k_bn_stats  <<<cdiv(N, 128), H, 0, stream>>>(t1, stats, N, H, 128);
  k_bn_apply  <<<cdiv(N * H, TPB), TPB, 0, stream>>>(t1, stats, g1, be1, t1, N, SH, 1);

  // ---- GCN layer 2: t0 = t1 @ W2 ; aggregate -> t1 ; BN+ReLU -> emb (d_out)
  { int tn = cdiv(H, 16), tiles = (N / 16) * tn;
    k_gemm_wmma<<<cdiv(tiles, 8), TPB, 0, stream>>>(t1, W2, nullptr, t0, N, H, H, tn); }
  k_zero      <<<cdiv(N * H, TPB), TPB, 0, stream>>>(t1, N * H);
  k_scatter128<<<4096, TPB, 0, stream>>>(t0, src, dst, dinv, t1, E);
  k_finalize  <<<cdiv(N * H, TPB), TPB, 0, stream>>>(t1, t0, dinv, b2, N, SH);
  k_zero      <<<1, 256, 0, stream>>>(stats, 2 * H);
  k_bn_stats  <<<cdiv(N, 128), H, 0, stream>>>(t1, stats, N, H, 128);
  k_bn_apply  <<<cdiv(N * H, TPB), TPB, 0, stream>>>(t1, stats, g2, be2, emb, N, SH, 1);

  // ---- forward head: 128 -> 128 -> 64 -> 50
  { int tn = cdiv(H, 16), tiles = (N / 16) * tn;
    k_gemm_wmma<<<cdiv(tiles, 8), TPB, 0, stream>>>(emb, fW1, fb1, t0, N, H, H, tn); }
  k_zero    <<<1, 256, 0, stream>>>(stats, 2 * H);
  k_bn_stats<<<cdiv(N, 128), H, 0, stream>>>(t0, stats, N, H, 128);
  k_bn_apply<<<cdiv(N * H, TPB), TPB, 0, stream>>>(t0, stats, fg1, fbe1, t0, N, SH, 1);

  { int tn = cdiv(H2, 16), tiles = (N / 16) * tn;
    k_gemm_wmma<<<cdiv(tiles, 8), TPB, 0, stream>>>(t0, fW2, fb2, t2, N, H, H2, tn); }
  k_zero    <<<1, 128, 0, stream>>>(stats, 2 * H2);
  k_bn_stats<<<cdiv(N, 128), H2, 0, stream>>>(t2, stats, N, H2, 128);
  k_bn_apply<<<cdiv(N * H2, TPB), TPB, 0, stream>>>(t2, stats, fg2, fbe2, t2, N, SH2, 1);

  { int tn = cdiv(NEX, 16), tiles = (N / 16) * tn;
    k_gemm_wmma<<<cdiv(tiles, 8), TPB, 0, stream>>>(t2, fW3, fb3, fwd, N, H2, NEX, tn); }

  // ---- backward head: 128 -> 64 -> 3
  { int tn = cdiv(H2, 16), tiles = (N / 16) * tn;
    k_gemm_wmma<<<cdiv(tiles, 8), TPB, 0, stream>>>(emb, bW1, bb1, t2, N, H, H2, tn); }
  k_zero    <<<1, 128, 0, stream>>>(stats, 2 * H2);
  k_bn_stats<<<cdiv(N, 128), H2, 0, stream>>>(t2, stats, N, H2, 128);
  k_bn_apply<<<cdiv(N * H2, TPB), TPB, 0, stream>>>(t2, stats, bg1, bbe1, t2, N, SH2, 1);

  { int tn = cdiv(NCAT, 16), tiles = (N / 16) * tn;
    k_gemm_wmma<<<cdiv(tiles, 8), TPB, 0, stream>>>(t2, bW2, bb2, bwd, N, H2, NCAT, tn); }
}